// NCC_3143916060729
// MI455X (gfx1250) — compile-verified
//
#include <hip/hip_runtime.h>
#include <hip/hip_bf16.h>

typedef __attribute__((ext_vector_type(2))) float v2f;
typedef __attribute__((ext_vector_type(8))) float v8f;

#define DVOL 160
#define TILE 16
#define PAD  4
#define REG  24   // TILE + 2*PAD

__device__ __forceinline__ v8f wmma_f32(v2f a, v2f b, v8f c) {
  // D(16x16,f32) = A(16x4,f32) * B(4x16,f32) + C
  return __builtin_amdgcn_wmma_f32_16x16x4_f32(
      /*neg_a=*/false, a, /*neg_b=*/false, b,
      /*c_mod=*/(short)0, c, /*reuse_a=*/false, /*reuse_b=*/false);
}

__global__ void __launch_bounds__(32) ncc_init(float* out) {
  if (threadIdx.x == 0) out[0] = 1.0f;
}

__global__ void __launch_bounds__(32) ncc_main(const float* __restrict__ I,
                                               const float* __restrict__ J,
                                               float* __restrict__ out) {
  // Per-block (one wave32) LDS: 63.3 KB total
  __shared__ float  lsI[32][25];        // 24x24 slice tile, rows 24..31 zero
  __shared__ float  lsJ[32][25];
  __shared__ float  tmpF[5][32][17];    // x-pass result per field (24x16 valid)
  __shared__ float4 ring[9][5][32][2];  // 9 z-slabs x 5 fields, D-fragment layout

  const int lane = threadIdx.x;   // 0..31
  const int m    = lane & 15;     // N index (x-pass) / M index (y-pass)
  const int half = lane >> 4;

  int id = blockIdx.x;
  const int tx = id % 10; id /= 10;
  const int ty = id % 10; id /= 10;
  const int tz = id % 10; id /= 10;
  const int b  = id;              // 0..1

  const int x0  = tx * TILE;
  const int y0  = ty * TILE;
  const int zo0 = tz * TILE;

  // Zero the padding rows 24..31 of the input tiles (read as zeros by WMMA).
  for (int idx = lane; idx < 8 * 25; idx += 32) {
    const int r = 24 + idx / 25, c = idx % 25;
    lsI[r][c] = 0.0f;
    lsJ[r][c] = 0.0f;
  }

  // Band-matrix fragments: band(k, j) = (j <= k <= j+8).
  // Serve as B in the x-pass (j = N) and as A in the y-pass (j = M);
  // per-lane layout is identical for both (k = 4c + 2*half + {0,1}, j = m).
  v2f bb[6];
#pragma unroll
  for (int c = 0; c < 6; ++c) {
    const int k0 = 4 * c + 2 * half;
    bb[c][0] = (m <= k0     && k0     <= m + 8) ? 1.0f : 0.0f;
    bb[c][1] = (m <= k0 + 1 && k0 + 1 <= m + 8) ? 1.0f : 0.0f;
  }

  float ccAcc = 0.0f;

  for (int step = 0; step < 24; ++step) {
    const int zi  = zo0 - PAD + step;       // input slice
    const bool zok = (zi >= 0) && (zi < DVOL);

    // ---- Load 24x24 halo slice of I and J (zero-padded) into LDS ----
#pragma unroll 6
    for (int idx = lane; idx < REG * REG; idx += 32) {
      const int r  = idx / REG, c = idx - r * REG;
      const int gy = y0 - PAD + r;
      const int gx = x0 - PAD + c;
      float vi = 0.0f, vj = 0.0f;
      if (zok && (unsigned)gy < (unsigned)DVOL && (unsigned)gx < (unsigned)DVOL) {
        const size_t g = ((size_t)(b * DVOL + zi) * DVOL + gy) * DVOL + gx;
        vi = I[g];
        vj = J[g];
      }
      lsI[r][c] = vi;
      lsJ[r][c] = vj;
    }
    // Prefetch next slice's first line into cache (CDNA5 global_prefetch_b8).
    if (zok && zi + 1 < DVOL) {
      const size_t gp = ((size_t)(b * DVOL + zi + 1) * DVOL + y0) * DVOL + x0;
      __builtin_prefetch(&I[gp], 0, 1);
      __builtin_prefetch(&J[gp], 0, 1);
    }
    asm volatile("s_wait_dscnt 0" ::: "memory");

    // ---- x-pass: tmp(24x16) = Region(24x24) x Band(24x16), 5 fields fused ----
#pragma unroll
    for (int h = 0; h < 2; ++h) {
      v8f aI = {}, aJ = {}, aII = {}, aJJ = {}, aIJ = {};
      const int row = h * 16 + m;
#pragma unroll
      for (int c = 0; c < 6; ++c) {
        const int c0 = 4 * c + 2 * half;
        const float i0 = lsI[row][c0], i1 = lsI[row][c0 + 1];
        const float j0 = lsJ[row][c0], j1 = lsJ[row][c0 + 1];
        const v2f vI  = {i0, i1};
        const v2f vJ  = {j0, j1};
        const v2f vII = {i0 * i0, i1 * i1};
        const v2f vJJ = {j0 * j0, j1 * j1};
        const v2f vIJ = {i0 * j0, i1 * j1};
        aI  = wmma_f32(vI,  bb[c], aI);
        aJ  = wmma_f32(vJ,  bb[c], aJ);
        aII = wmma_f32(vII, bb[c], aII);
        aJJ = wmma_f32(vJJ, bb[c], aJJ);
        aIJ = wmma_f32(vIJ, bb[c], aIJ);
      }
#pragma unroll
      for (int r = 0; r < 8; ++r) {
        const int rt = h * 16 + r + 8 * half;  // D layout: M = r + 8*half
        tmpF[0][rt][m] = aI[r];
        tmpF[1][rt][m] = aJ[r];
        tmpF[2][rt][m] = aII[r];
        tmpF[3][rt][m] = aJJ[r];
        tmpF[4][rt][m] = aIJ[r];
      }
    }
    asm volatile("s_wait_dscnt 0" ::: "memory");

    // ---- y-pass: Out(16x16) = Band(16x24) x tmp(24x16); stash in ring ----
    const int s = step % 9;
#pragma unroll
    for (int f = 0; f < 5; ++f) {
      v8f acc = {};
#pragma unroll
      for (int c = 0; c < 6; ++c) {
        const int k0 = 4 * c + 2 * half;
        const v2f vB = {tmpF[f][k0][m], tmpF[f][k0 + 1][m]};
        acc = wmma_f32(bb[c], vB, acc);
      }
      ring[s][f][lane][0] = make_float4(acc[0], acc[1], acc[2], acc[3]);
      ring[s][f][lane][1] = make_float4(acc[4], acc[5], acc[6], acc[7]);
    }
    asm volatile("s_wait_dscnt 0" ::: "memory");

    // ---- When 9 slabs are live, emit output slice zo = zi - 4 ----
    if (step >= 8) {
      float sums[5][8];
#pragma unroll
      for (int f = 0; f < 5; ++f)
#pragma unroll
        for (int r = 0; r < 8; ++r) sums[f][r] = 0.0f;

#pragma unroll
      for (int sl = 0; sl < 9; ++sl)
#pragma unroll
        for (int f = 0; f < 5; ++f) {
          const float4 p0 = ring[sl][f][lane][0];
          const float4 p1 = ring[sl][f][lane][1];
          sums[f][0] += p0.x; sums[f][1] += p0.y;
          sums[f][2] += p0.z; sums[f][3] += p0.w;
          sums[f][4] += p1.x; sums[f][5] += p1.y;
          sums[f][6] += p1.z; sums[f][7] += p1.w;
        }

      const float win    = 729.0f;          // 9^3
      const float invWin = 1.0f / 729.0f;
#pragma unroll
      for (int r = 0; r < 8; ++r) {
        const float Is = sums[0][r], Js = sums[1][r];
        const float I2 = sums[2][r], J2 = sums[3][r], IJ = sums[4][r];
        const float uI = Is * invWin, uJ = Js * invWin;
        const float cross = IJ - uJ * Is - uI * Js + uI * uJ * win;
        const float Iv = I2 - 2.0f * uI * Is + uI * uI * win;
        const float Jv = J2 - 2.0f * uJ * Js + uJ * uJ * win;
        float cc = (cross * cross) / (Iv * Jv + 1e-5f);
        cc = fminf(fmaxf(cc, 0.0f), 1.0f);
        ccAcc += cc;
      }
    }
  }

  // Wave reduction + single atomic: out = 1 - sum(cc)/N
#pragma unroll
  for (int off = 16; off >= 1; off >>= 1)
    ccAcc += __shfl_xor(ccAcc, off, 32);
  if (lane == 0)
    atomicAdd(out, -ccAcc * (1.0f / 8192000.0f));  // N = 2*160^3
}

extern "C" void kernel_launch(void* const* d_in, const int* in_sizes, int n_in,
                              void* d_out, int out_size, void* d_ws, size_t ws_size,
                              hipStream_t stream) {
  const float* I = (const float*)d_in[0];  // predicted
  const float* J = (const float*)d_in[1];  // target
  float* out = (float*)d_out;

  ncc_init<<<dim3(1), dim3(32), 0, stream>>>(out);
  // 2 batches x 10 z-segments x 10x10 (y,x) tiles = 2000 single-wave blocks
  ncc_main<<<dim3(2000), dim3(32), 0, stream>>>(I, J, out);
}